// ProbAttention_83485574300199
// MI455X (gfx1250) — compile-verified
//
#include <hip/hip_runtime.h>
#include <math.h>

#define T_DIM 2048
#define C_DIM 512
#define H_DIM 8
#define D_DIM 64
#define NTOP 40
#define SKEY 40

typedef __attribute__((ext_vector_type(2))) float v2f;
typedef __attribute__((ext_vector_type(8))) float v8f;

// ---------------------------------------------------------------------------
// GEMM: Y[2048,512] = A[2048,512] @ W[512,512] + bias[512]
// fp32 WMMA 16x16x4, one 16x16 D-tile per wave, 8 waves per block.
// grid = (T/16, (C/16)/8) = (128, 4), block = 256.
// ---------------------------------------------------------------------------
__global__ __launch_bounds__(256)
void pa_gemm_f32_wmma(const float* __restrict__ A, const float* __restrict__ W,
                      const float* __restrict__ bias, float* __restrict__ Y) {
  const int lane = threadIdx.x & 31;
  const int wave = threadIdx.x >> 5;
  const int row0 = blockIdx.x * 16;
  const int col0 = (blockIdx.y * 8 + wave) * 16;
  const int r  = lane & 15;   // A row / B,D column within tile
  const int kh = lane >> 4;   // K-half select (0/1)

  v8f acc = {};
  const float* __restrict__ arow = A + (size_t)(row0 + r) * C_DIM;
  const float* __restrict__ bcol = W + (col0 + r);

  #pragma unroll 4
  for (int k0 = 0; k0 < C_DIM; k0 += 4) {
    const int ka = k0 + 2 * kh;                  // even -> 8B aligned
    v2f a = *(const v2f*)(arow + ka);            // A[row][ka], A[row][ka+1]
    v2f b;
    b.x = bcol[(size_t)ka * C_DIM];              // W[ka][col]
    b.y = bcol[(size_t)(ka + 1) * C_DIM];        // W[ka+1][col]
    acc = __builtin_amdgcn_wmma_f32_16x16x4_f32(false, a, false, b,
                                                (short)0, acc, false, false);
  }

  const float bv = bias[col0 + r];
  #pragma unroll
  for (int g = 0; g < 8; ++g)
    Y[(size_t)(row0 + g + 8 * kh) * C_DIM + col0 + r] = acc[g] + bv;
}

// ---------------------------------------------------------------------------
// M[h][t] = max_s(q[t]·k[idx[t,s]]) - sum_s(q[t]·k[idx[t,s]]) / T
// one thread per (h,t); H*T = 16384 threads.
// ---------------------------------------------------------------------------
__global__ __launch_bounds__(256)
void pa_sample_scores(const float* __restrict__ qb, const float* __restrict__ kb,
                      const int* __restrict__ idxs, float* __restrict__ M) {
  const int gid = blockIdx.x * 256 + threadIdx.x;
  if (gid >= H_DIM * T_DIM) return;
  const int h = gid / T_DIM;
  const int t = gid % T_DIM;

  const float* __restrict__ qrow = qb + (size_t)t * C_DIM + h * D_DIM;
  float q[D_DIM];
  #pragma unroll
  for (int d = 0; d < D_DIM; ++d) q[d] = qrow[d];

  float mx = -INFINITY, sum = 0.f;
  for (int s = 0; s < SKEY; ++s) {
    const int j = idxs[t * SKEY + s];
    const float* __restrict__ krow = kb + (size_t)j * C_DIM + h * D_DIM;
    float acc = 0.f;
    #pragma unroll
    for (int d = 0; d < D_DIM; ++d) acc += q[d] * krow[d];
    mx = fmaxf(mx, acc);
    sum += acc;
  }
  M[h * T_DIM + t] = mx - sum * (1.0f / (float)T_DIM);
}

// ---------------------------------------------------------------------------
// Per-head iterative top-40 argmax (ties -> lowest index). grid = H, block=256.
// ---------------------------------------------------------------------------
__global__ __launch_bounds__(256)
void pa_topk(const float* __restrict__ M, int* __restrict__ Mtop) {
  const int h = blockIdx.x;
  const int tid = threadIdx.x;
  __shared__ float sm[T_DIM];
  __shared__ float rv[256];
  __shared__ int   ri[256];

  for (int i = tid; i < T_DIM; i += 256) sm[i] = M[h * T_DIM + i];
  __syncthreads();

  for (int it = 0; it < NTOP; ++it) {
    float bvv = -INFINITY; int bii = T_DIM;
    for (int i = tid; i < T_DIM; i += 256) {
      const float v = sm[i];
      if (v > bvv || (v == bvv && i < bii)) { bvv = v; bii = i; }
    }
    rv[tid] = bvv; ri[tid] = bii;
    __syncthreads();
    for (int s = 128; s > 0; s >>= 1) {
      if (tid < s) {
        const float ov = rv[tid + s]; const int oi = ri[tid + s];
        if (ov > rv[tid] || (ov == rv[tid] && oi < ri[tid])) { rv[tid] = ov; ri[tid] = oi; }
      }
      __syncthreads();
    }
    if (tid == 0) { Mtop[h * NTOP + it] = ri[0]; sm[ri[0]] = -INFINITY; }
    __syncthreads();
  }
}

// ---------------------------------------------------------------------------
// Sparse attention: per (h,u) -> one block. scores over 2048 keys, causal
// mask, softmax, upd[h][u][:] = attn @ V. block = 256.
// ---------------------------------------------------------------------------
__global__ __launch_bounds__(256)
void pa_sparse_attn(const float* __restrict__ qb, const float* __restrict__ kb,
                    const float* __restrict__ vb, const int* __restrict__ Mtop,
                    float* __restrict__ upd) {
  const int h = blockIdx.x, u = blockIdx.y;
  const int tid = threadIdx.x;
  const int pos = Mtop[h * NTOP + u];

  __shared__ float p[T_DIM];
  __shared__ float red[256];
  __shared__ float qs[D_DIM];
  __shared__ float pacc[4][D_DIM];

  if (tid < D_DIM) qs[tid] = qb[(size_t)pos * C_DIM + h * D_DIM + tid];
  __syncthreads();

  const float scale = 0.125f;  // 1/sqrt(64)
  float lmax = -INFINITY;
  for (int t = tid; t < T_DIM; t += 256) {
    float s;
    if (t > pos) {
      s = -INFINITY;
    } else {
      const float* __restrict__ krow = kb + (size_t)t * C_DIM + h * D_DIM;
      float acc = 0.f;
      #pragma unroll
      for (int d = 0; d < D_DIM; ++d) acc += qs[d] * krow[d];
      s = acc * scale;
    }
    p[t] = s;
    lmax = fmaxf(lmax, s);
  }
  red[tid] = lmax; __syncthreads();
  for (int s = 128; s > 0; s >>= 1) { if (tid < s) red[tid] = fmaxf(red[tid], red[tid + s]); __syncthreads(); }
  const float m = red[0];
  __syncthreads();

  float lsum = 0.f;
  for (int t = tid; t < T_DIM; t += 256) {
    const float e = (t > pos) ? 0.f : expf(p[t] - m);
    p[t] = e;
    lsum += e;
  }
  red[tid] = lsum; __syncthreads();
  for (int s = 128; s > 0; s >>= 1) { if (tid < s) red[tid] += red[tid + s]; __syncthreads(); }
  const float linv = 1.0f / red[0];
  __syncthreads();

  const int d = tid & 63, chunk = tid >> 6;
  float acc = 0.f;
  const int t0 = chunk * (T_DIM / 4), t1 = t0 + (T_DIM / 4);
  for (int t = t0; t < t1; ++t)
    acc += p[t] * vb[(size_t)t * C_DIM + h * D_DIM + d];
  pacc[chunk][d] = acc;
  __syncthreads();
  if (tid < D_DIM) {
    const float s = pacc[0][tid] + pacc[1][tid] + pacc[2][tid] + pacc[3][tid];
    upd[(h * NTOP + u) * D_DIM + tid] = s * linv;
  }
}

// ---------------------------------------------------------------------------
// ctx[t][c] = inclusive cumsum over t of v[t][c]. One block per column c.
// block = 256, each thread handles 8 consecutive t.
// ---------------------------------------------------------------------------
__global__ __launch_bounds__(256)
void pa_cumsum(const float* __restrict__ vb, float* __restrict__ ctx) {
  const int c = blockIdx.x;
  const int tid = threadIdx.x;
  const int PT = T_DIM / 256;  // 8
  float vals[T_DIM / 256];
  float run = 0.f;
  #pragma unroll
  for (int i = 0; i < PT; ++i) {
    run += vb[(size_t)(tid * PT + i) * C_DIM + c];
    vals[i] = run;                           // local inclusive prefix
  }
  __shared__ float s[256];
  s[tid] = run; __syncthreads();
  for (int off = 1; off < 256; off <<= 1) {  // Hillis-Steele inclusive scan
    const float add = (tid >= off) ? s[tid - off] : 0.f;
    __syncthreads();
    s[tid] += add;
    __syncthreads();
  }
  const float excl = s[tid] - run;
  #pragma unroll
  for (int i = 0; i < PT; ++i)
    ctx[(size_t)(tid * PT + i) * C_DIM + c] = excl + vals[i];
}

// ---------------------------------------------------------------------------
// Scatter: ctx[Mtop[h][u]][h*D+d] = upd[h][u][d]
// ---------------------------------------------------------------------------
__global__ __launch_bounds__(256)
void pa_scatter(const int* __restrict__ Mtop, const float* __restrict__ upd,
                float* __restrict__ ctx) {
  const int idx = blockIdx.x * 256 + threadIdx.x;
  if (idx >= H_DIM * NTOP * D_DIM) return;
  const int d = idx % D_DIM;
  const int hu = idx / D_DIM;
  const int u = hu % NTOP;
  const int h = hu / NTOP;
  const int pos = Mtop[h * NTOP + u];
  ctx[(size_t)pos * C_DIM + h * D_DIM + d] = upd[idx];
}

// ---------------------------------------------------------------------------
extern "C" void kernel_launch(void* const* d_in, const int* in_sizes, int n_in,
                              void* d_out, int out_size, void* d_ws, size_t ws_size,
                              hipStream_t stream) {
  const float* x   = (const float*)d_in[0];
  const float* Wq  = (const float*)d_in[1];
  const float* bq  = (const float*)d_in[2];
  const float* Wk  = (const float*)d_in[3];
  const float* bk  = (const float*)d_in[4];
  const float* Wv  = (const float*)d_in[5];
  const float* bv  = (const float*)d_in[6];
  const float* Wp  = (const float*)d_in[7];
  const float* bp  = (const float*)d_in[8];
  const int*   smp = (const int*)d_in[9];

  const size_t TC = (size_t)T_DIM * C_DIM;
  float* qb   = (float*)d_ws;
  float* kb   = qb + TC;
  float* vb   = kb + TC;
  float* ctx  = vb + TC;
  float* M    = ctx + TC;
  int*   Mtop = (int*)(M + H_DIM * T_DIM);
  float* upd  = (float*)(Mtop + H_DIM * NTOP);

  dim3 gemm_grid(T_DIM / 16, (C_DIM / 16) / 8);   // (128, 4)

  // Q/K/V projections (fp32 WMMA)
  pa_gemm_f32_wmma<<<gemm_grid, 256, 0, stream>>>(x, Wq, bq, qb);
  pa_gemm_f32_wmma<<<gemm_grid, 256, 0, stream>>>(x, Wk, bk, kb);
  pa_gemm_f32_wmma<<<gemm_grid, 256, 0, stream>>>(x, Wv, bv, vb);

  // sampled sparsity measure
  pa_sample_scores<<<(H_DIM * T_DIM + 255) / 256, 256, 0, stream>>>(qb, kb, smp, M);

  // per-head top-40
  pa_topk<<<H_DIM, 256, 0, stream>>>(M, Mtop);

  // sparse attention rows
  pa_sparse_attn<<<dim3(H_DIM, NTOP), 256, 0, stream>>>(qb, kb, vb, Mtop, upd);

  // context = cumsum(v), then scatter attention rows
  pa_cumsum<<<C_DIM, 256, 0, stream>>>(vb, ctx);
  pa_scatter<<<(H_DIM * NTOP * D_DIM + 255) / 256, 256, 0, stream>>>(Mtop, upd, ctx);

  // output projection (fp32 WMMA)
  pa_gemm_f32_wmma<<<gemm_grid, 256, 0, stream>>>(ctx, Wp, bp, (float*)d_out);
}